// TargetPrediction_14791867368116
// MI455X (gfx1250) — compile-verified
//
#include <hip/hip_runtime.h>
#include <math.h>

#define BATCH 512
#define MPTS  2000
#define NSEL  1000
#define OUTN  50
#define HID   64
#define PI_F  3.14159265358979f
#define SMALLF 1e-06f

typedef float v2f __attribute__((ext_vector_type(2)));
typedef float v8f __attribute__((ext_vector_type(8)));
typedef int   v4i __attribute__((vector_size(16)));

__device__ __forceinline__ float gelu_exact(float x) {
    return 0.5f * x * (1.0f + erff(x * 0.70710678118654752f));
}
__device__ __forceinline__ float sl1(float d) {
    d = fabsf(d);
    return d < 1.0f ? 0.5f * d * d : d - 0.5f;
}
__device__ __forceinline__ float rsum8(float v) {
    v += __shfl_xor(v, 1);
    v += __shfl_xor(v, 2);
    v += __shfl_xor(v, 4);
    return v;
}

// -------- CDNA5 async global->LDS copy (ASYNCcnt-tracked), 16B granule --------
__device__ __forceinline__ void async_copy_b128(const float* g, float* l) {
#if __has_builtin(__builtin_amdgcn_global_load_async_to_lds_b128)
    __builtin_amdgcn_global_load_async_to_lds_b128(
        (__attribute__((address_space(1))) v4i*)g,
        (__attribute__((address_space(3))) v4i*)l, 0, 0);
#else
    unsigned lo = (unsigned)(size_t)(__attribute__((address_space(3))) void*)l;
    asm volatile("global_load_async_to_lds_b128 %0, %1, off"
                 :: "v"(lo), "v"(g) : "memory");
#endif
}
__device__ __forceinline__ void wait_async0() {
#if __has_builtin(__builtin_amdgcn_s_wait_asynccnt)
    __builtin_amdgcn_s_wait_asynccnt(0);
#else
    asm volatile("s_wait_asynccnt 0x0" ::: "memory");
#endif
}

// ---------------------------------------------------------------- init
__global__ void k_init(float* stats, float* accum) {
    int i = blockIdx.x * blockDim.x + threadIdx.x;
    if (i < BATCH * 4) stats[i] = 0.0f;
    if (i < 8) accum[i] = 0.0f;
}

// ---------------------------------------------------------------- select: top-1000 nearest + argmin to end
__global__ __launch_bounds__(1024) void k_select(const float* __restrict__ osm,
                                                 const float* __restrict__ endp,
                                                 float* __restrict__ pts,
                                                 int* __restrict__ tindex) {
    __shared__ float sd[2048];
    __shared__ int   si[2048];
    __shared__ unsigned long long s64[1024];
    const int b = blockIdx.x;
    const int tid = threadIdx.x;
    const float* obase = osm + (size_t)b * MPTS * 4;

    __builtin_prefetch(obase + tid * 8, 0, 0);

    for (int e = tid; e < 2048; e += 1024) {
        float d;
        if (e < MPTS) {
            float x = obase[e * 4 + 0];
            float y = obase[e * 4 + 1];
            d = x * x + y * y;   // monotone with sqrt: same ordering
        } else {
            d = INFINITY;
        }
        sd[e] = d;
        si[e] = e;
    }
    __syncthreads();

    // ascending bitonic sort of 2048 (dist, idx)
    for (int k = 2; k <= 2048; k <<= 1) {
        for (int j = k >> 1; j > 0; j >>= 1) {
            for (int e = tid; e < 2048; e += 1024) {
                int ixj = e ^ j;
                if (ixj > e) {
                    bool up = ((e & k) == 0);
                    float d0 = sd[e], d1 = sd[ixj];
                    if ((d0 > d1) == up) {
                        sd[e] = d1; sd[ixj] = d0;
                        int t = si[e]; si[e] = si[ixj]; si[ixj] = t;
                    }
                }
            }
            __syncthreads();
        }
    }

    const float ex = endp[b * 2 + 0], ey = endp[b * 2 + 1];
    unsigned long long best = ~0ull;
    for (int e = tid; e < NSEL; e += 1024) {
        int src = si[e];
        float x = obase[src * 4 + 0];
        float y = obase[src * 4 + 1];
        pts[((size_t)b * NSEL + e) * 2 + 0] = x;
        pts[((size_t)b * NSEL + e) * 2 + 1] = y;
        float dx = x - ex, dy = y - ey;
        float dd = dx * dx + dy * dy;  // argmin(dist) == argmax(softmax(-dist/T))
        unsigned long long key =
            (((unsigned long long)__float_as_uint(dd)) << 32) | (unsigned)e;
        best = key < best ? key : best;
    }
    s64[tid] = best;
    __syncthreads();
    for (int s = 512; s > 0; s >>= 1) {
        if (tid < s) {
            unsigned long long o = s64[tid + s];
            if (o < s64[tid]) s64[tid] = o;
        }
        __syncthreads();
    }
    if (tid == 0) tindex[b] = (int)(s64[0] & 0xffffffffull);
}

// ---------------------------------------------------------------- base: fold constant (vf,at) columns through W1
__global__ __launch_bounds__(64) void k_base(const float* __restrict__ vf,
                                             const float* __restrict__ at,
                                             const float* __restrict__ oxw1, const float* __restrict__ oxb1,
                                             const float* __restrict__ tpw1, const float* __restrict__ tpb1,
                                             float* __restrict__ base_ox, float* __restrict__ base_tp) {
    int b = blockIdx.x, j = threadIdx.x;
    const float* v = vf + b * 64;
    const float* a = at + b * 64;
    float so = oxb1[j], st = tpb1[j];
    for (int i = 0; i < 64; ++i) {
        float vi = v[i], ai = a[i];
        so += vi * oxw1[i * HID + j] + ai * oxw1[(64 + i) * HID + j];
        st += vi * tpw1[i * HID + j] + ai * tpw1[(64 + i) * HID + j];
    }
    base_ox[b * 64 + j] = so;
    base_tp[b * 64 + j] = st;
}

// ---------------------------------------------------------------- MLP with f32 WMMA for h @ tp_w2
#define W2S 72   // sW2 row stride: 288B rows -> 16B aligned, lane-halves hit disjoint banks

__global__ __launch_bounds__(128) void k_mlp(
    const float* __restrict__ pts, const float* __restrict__ endp,
    const float* __restrict__ base_ox,
    const float* __restrict__ oxw1, const float* __restrict__ oxg1, const float* __restrict__ oxbe1,
    const float* __restrict__ oxw2, const float* __restrict__ oxb2,
    const float* __restrict__ base_tp,
    const float* __restrict__ tpw1, const float* __restrict__ tpg1, const float* __restrict__ tpbe1,
    const float* __restrict__ tpw2, const float* __restrict__ tpb2,
    const float* __restrict__ tpg2, const float* __restrict__ tpbe2,
    const float* __restrict__ tpw3, const float* __restrict__ tpb3,
    float* __restrict__ targets, float* __restrict__ stats, float* __restrict__ accum) {

    __shared__ float sW2[64][W2S];  // tp_w2 staged via async global->LDS DMA
    __shared__ float sA[16][65];    // 16-row activation tile

    const int b = blockIdx.y;
    const int row0 = blockIdx.x * 16;
    const int tid = threadIdx.x;
    const int r = tid >> 3;        // row within tile (0..15)
    const int c = tid & 7;         // 8 lanes per row
    const int f0 = c * 8;          // each lane owns 8 features
    const int row = row0 + r;
    const bool valid = row < NSEL;

    // async-stage tp_w2: 1024 x b128 chunks, 8 per thread; overlaps the math below
    for (int q = tid; q < 64 * 16; q += 128) {
        int wr = q >> 4;
        int wc = (q & 15) * 4;
        async_copy_b128(tpw2 + wr * 64 + wc, &sW2[wr][wc]);
    }

    float px = 0.0f, py = 0.0f;
    if (valid) {
        px = pts[((size_t)b * NSEL + row) * 2 + 0];
        py = pts[((size_t)b * NSEL + row) * 2 + 1];
    }

    float h[8];

    // ---- offset MLP (ox): rank-2 update + LN + GELU + 64->2 projection ----
    {
        float s = 0.0f;
        for (int j = 0; j < 8; ++j) {
            int f = f0 + j;
            h[j] = base_ox[b * 64 + f] + px * oxw1[128 * HID + f] + py * oxw1[129 * HID + f];
            s += h[j];
        }
        float mu = rsum8(s) * (1.0f / 64.0f);
        float q = 0.0f;
        for (int j = 0; j < 8; ++j) { h[j] -= mu; q += h[j] * h[j]; }
        float inv = rsqrtf(rsum8(q) * (1.0f / 64.0f) + 1e-5f);
        float ox = 0.0f, oy = 0.0f;
        for (int j = 0; j < 8; ++j) {
            int f = f0 + j;
            float g = gelu_exact(h[j] * inv * oxg1[f] + oxbe1[f]);
            ox += g * oxw2[f * 2 + 0];
            oy += g * oxw2[f * 2 + 1];
        }
        px += rsum8(ox) + oxb2[0];   // intermediate targets = pts + offset
        py += rsum8(oy) + oxb2[1];
    }

    // ---- tp layer 1: rank-2 update + LN + GELU -> LDS tile ----
    {
        float s = 0.0f;
        for (int j = 0; j < 8; ++j) {
            int f = f0 + j;
            h[j] = base_tp[b * 64 + f] + px * tpw1[128 * HID + f] + py * tpw1[129 * HID + f];
            s += h[j];
        }
        float mu = rsum8(s) * (1.0f / 64.0f);
        float q = 0.0f;
        for (int j = 0; j < 8; ++j) { h[j] -= mu; q += h[j] * h[j]; }
        float inv = rsqrtf(rsum8(q) * (1.0f / 64.0f) + 1e-5f);
        for (int j = 0; j < 8; ++j) {
            int f = f0 + j;
            sA[r][f] = gelu_exact(h[j] * inv * tpg1[f] + tpbe1[f]);
        }
    }
    wait_async0();                 // sW2 DMA complete (ASYNCcnt==0)
    __syncthreads();

    // ---- 16x64 @ 64x64 GEMM: each wave owns one 16x16 N-tile, 16 x WMMA f32 16x16x4 ----
    const int wv = tid >> 5;       // N tile index (0..3), 4 waves
    const int lane = tid & 31;
    const int m = lane & 15;       // A: row M / B,D: column N
    const int kh = lane >> 4;      // lane-half selects K pair (0,1) vs (2,3)

    v8f acc;
    {
        float bias = tpb2[wv * 16 + m];
        for (int v = 0; v < 8; ++v) acc[v] = bias;
    }
    for (int k = 0; k < 64; k += 4) {
        v2f a, bb;
        a.x  = sA[m][k + 2 * kh + 0];
        a.y  = sA[m][k + 2 * kh + 1];
        bb.x = sW2[k + 2 * kh + 0][wv * 16 + m];
        bb.y = sW2[k + 2 * kh + 1][wv * 16 + m];
        acc = __builtin_amdgcn_wmma_f32_16x16x4_f32(false, a, false, bb,
                                                    (short)0, acc, false, false);
    }
    __syncthreads();               // everyone done reading sA before overwrite
    for (int v = 0; v < 8; ++v)    // C/D layout: VGPR v -> M = v + 8*lane_half
        sA[v + 8 * kh][wv * 16 + m] = acc[v];
    __syncthreads();

    // ---- LN2 + GELU + 64->2 projection ----
    {
        float s = 0.0f;
        for (int j = 0; j < 8; ++j) { h[j] = sA[r][f0 + j]; s += h[j]; }
        float mu = rsum8(s) * (1.0f / 64.0f);
        float q = 0.0f;
        for (int j = 0; j < 8; ++j) { h[j] -= mu; q += h[j] * h[j]; }
        float inv = rsqrtf(rsum8(q) * (1.0f / 64.0f) + 1e-5f);
        float fx = 0.0f, fy = 0.0f;
        for (int j = 0; j < 8; ++j) {
            int f = f0 + j;
            float g = gelu_exact(h[j] * inv * tpg2[f] + tpbe2[f]);
            fx += g * tpw3[f * 2 + 0];
            fy += g * tpw3[f * 2 + 1];
        }
        fx = rsum8(fx) + tpb3[0];
        fy = rsum8(fy) + tpb3[1];

        if (valid && c == 0) {
            targets[((size_t)b * NSEL + row) * 2 + 0] = fx;
            targets[((size_t)b * NSEL + row) * 2 + 1] = fy;
            atomicAdd(&stats[b * 4 + 0], fx);
            atomicAdd(&stats[b * 4 + 1], fy);
            atomicAdd(&stats[b * 4 + 2], fx * fx);
            atomicAdd(&stats[b * 4 + 3], fy * fy);
            float l = sl1(fx - endp[b * 2 + 0]) + sl1(fy - endp[b * 2 + 1]);
            atomicAdd(&accum[0], l);    // L_off numerator
        }
    }
}

// ---------------------------------------------------------------- policy density, L_cls, top-50
__global__ __launch_bounds__(512) void k_policy(const float* __restrict__ targets,
                                                const float* __restrict__ endp,
                                                const float* __restrict__ stats,
                                                const int* __restrict__ tindex,
                                                float* __restrict__ out,
                                                float* __restrict__ accum) {
    __shared__ float sp[1024];
    __shared__ int   si[1024];
    __shared__ float red[512];
    const int b = blockIdx.x;
    const int tid = threadIdx.x;

    const float n = (float)NSEL;
    float mx = stats[b * 4 + 0] / n, my = stats[b * 4 + 1] / n;
    float vx = (stats[b * 4 + 2] - n * mx * mx) / (n - 1.0f);
    float vy = (stats[b * 4 + 3] - n * my * my) / (n - 1.0f);
    float cx = 1.0f / sqrtf(2.0f * PI_F * vx + SMALLF);
    float cy = 1.0f / sqrtf(2.0f * PI_F * vy + SMALLF);

    for (int e = tid; e < 1024; e += 512) {
        float p;
        if (e < NSEL) {
            float tx = targets[((size_t)b * NSEL + e) * 2 + 0];
            float ty = targets[((size_t)b * NSEL + e) * 2 + 1];
            float pdx = expf(-0.5f * (tx - mx) * (tx - mx) / vx + SMALLF) * cx;
            float pdy = expf(-0.5f * (ty - my) * (ty - my) / vy + SMALLF) * cy;
            p = pdx * pdy;
        } else {
            p = -INFINITY;
        }
        sp[e] = p;
        si[e] = e;
    }
    __syncthreads();

    // logsumexp over policy values (e < NSEL)
    float lm = -INFINITY;
    for (int e = tid; e < NSEL; e += 512) lm = fmaxf(lm, sp[e]);
    red[tid] = lm;
    __syncthreads();
    for (int s = 256; s > 0; s >>= 1) {
        if (tid < s) red[tid] = fmaxf(red[tid], red[tid + s]);
        __syncthreads();
    }
    float pmax = red[0];
    __syncthreads();
    float ls = 0.0f;
    for (int e = tid; e < NSEL; e += 512) ls += expf(sp[e] - pmax);
    red[tid] = ls;
    __syncthreads();
    for (int s = 256; s > 0; s >>= 1) {
        if (tid < s) red[tid] += red[tid + s];
        __syncthreads();
    }
    float lse = pmax + logf(red[0]);
    float picked = sp[tindex[b]];
    if (tid == 0) atomicAdd(&accum[1], lse - picked);   // L_cls numerator
    __syncthreads();

    // descending bitonic sort of 1024 by policy value
    for (int k = 2; k <= 1024; k <<= 1) {
        for (int j = k >> 1; j > 0; j >>= 1) {
            for (int e = tid; e < 1024; e += 512) {
                int ixj = e ^ j;
                if (ixj > e) {
                    bool up = ((e & k) == 0);
                    float a = sp[e], bb = sp[ixj];
                    if ((a < bb) == up) {
                        sp[e] = bb; sp[ixj] = a;
                        int t = si[e]; si[e] = si[ixj]; si[ixj] = t;
                    }
                }
            }
            __syncthreads();
        }
    }

    if (tid < OUTN) {
        int src = si[tid];
        float tx = targets[((size_t)b * NSEL + src) * 2 + 0];
        float ty = targets[((size_t)b * NSEL + src) * 2 + 1];
        out[2 + ((size_t)b * OUTN + tid) * 2 + 0] = tx;
        out[2 + ((size_t)b * OUTN + tid) * 2 + 1] = ty;
        float l = sl1(tx - endp[b * 2 + 0]) + sl1(ty - endp[b * 2 + 1]);
        atomicAdd(&accum[2], l);    // L_end numerator
    }
}

// ---------------------------------------------------------------- finalize scalars
__global__ void k_final(const float* __restrict__ accum, float* __restrict__ out) {
    out[0] = accum[1] / (float)BATCH;
    out[1] = accum[0] / ((float)BATCH * NSEL * 2.0f)
           + 3.0f * accum[2] / ((float)BATCH * OUTN * 2.0f);
}

// ----------------------------------------------------------------
extern "C" void kernel_launch(void* const* d_in, const int* in_sizes, int n_in,
                              void* d_out, int out_size, void* d_ws, size_t ws_size,
                              hipStream_t stream) {
    (void)in_sizes; (void)n_in; (void)out_size; (void)ws_size;
    const float* osm   = (const float*)d_in[0];
    const float* atrj  = (const float*)d_in[1];
    const float* vf    = (const float*)d_in[2];
    const float* endp  = (const float*)d_in[3];
    /* d_in[4] = speed (unused by reference) */
    const float* oxw1  = (const float*)d_in[5];
    const float* oxb1  = (const float*)d_in[6];
    const float* oxg1  = (const float*)d_in[7];
    const float* oxbe1 = (const float*)d_in[8];
    const float* oxw2  = (const float*)d_in[9];
    const float* oxb2  = (const float*)d_in[10];
    const float* tpw1  = (const float*)d_in[11];
    const float* tpb1  = (const float*)d_in[12];
    const float* tpg1  = (const float*)d_in[13];
    const float* tpbe1 = (const float*)d_in[14];
    const float* tpw2  = (const float*)d_in[15];
    const float* tpb2  = (const float*)d_in[16];
    const float* tpg2  = (const float*)d_in[17];
    const float* tpbe2 = (const float*)d_in[18];
    const float* tpw3  = (const float*)d_in[19];
    const float* tpb3  = (const float*)d_in[20];

    float* ws      = (float*)d_ws;
    float* pts     = ws;                               // 512*1000*2
    float* targets = pts + (size_t)BATCH * NSEL * 2;   // 512*1000*2
    float* base_ox = targets + (size_t)BATCH * NSEL * 2;
    float* base_tp = base_ox + BATCH * 64;
    float* stats   = base_tp + BATCH * 64;             // 512*4
    float* accum   = stats + BATCH * 4;                // 8 floats
    int*   tindex  = (int*)(accum + 8);                // 512 ints
    float* out     = (float*)d_out;

    k_init<<<dim3((BATCH * 4 + 255) / 256), dim3(256), 0, stream>>>(stats, accum);
    k_select<<<dim3(BATCH), dim3(1024), 0, stream>>>(osm, endp, pts, tindex);
    k_base<<<dim3(BATCH), dim3(64), 0, stream>>>(vf, atrj, oxw1, oxb1, tpw1, tpb1,
                                                 base_ox, base_tp);
    k_mlp<<<dim3(64, BATCH), dim3(128), 0, stream>>>(
        pts, endp,
        base_ox, oxw1, oxg1, oxbe1, oxw2, oxb2,
        base_tp, tpw1, tpg1, tpbe1, tpw2, tpb2, tpg2, tpbe2, tpw3, tpb3,
        targets, stats, accum);
    k_policy<<<dim3(BATCH), dim3(512), 0, stream>>>(targets, endp, stats, tindex,
                                                    out, accum);
    k_final<<<1, 1, 0, stream>>>(accum, out);
}